// AdaptiveScaleVolume_28415503630947
// MI455X (gfx1250) — compile-verified
//
#include <hip/hip_runtime.h>
#include <hip/hip_bf16.h>

// ---------------------------------------------------------------------------
// AdaptiveScaleVolume for MI455X (gfx1250, wave32, WMMA + async LDS loads)
// B=2 C=64 H=96 W=192 D=48 G=8 GC=8 OUT=32
// Main compute: 3 dilated 3x3x3 convs folded with the final 1x1x1 conv into
// fp32 WMMA GEMMs (M=32, K=216) over the L2-resident correlation volume,
// with the im2col B-panel double-buffered in LDS via async global->LDS loads.
// ---------------------------------------------------------------------------

#define Bb 2
#define Cc 64
#define Hh 96
#define Ww 192
#define Dd 48
#define Gg 8
#define GC 8
#define OUT 32
#define KTOT 216        // G * 27 taps
#define PSTR 72         // LDS panel row stride (dwords): 2*72 % 64 == 16 -> no bank conflicts
#define PANSZ (KTOT * PSTR)

typedef __attribute__((ext_vector_type(2))) float v2f;
typedef __attribute__((ext_vector_type(8))) float v8f;

#if __has_builtin(__builtin_amdgcn_global_load_async_to_lds_b32) && \
    __has_builtin(__builtin_amdgcn_s_wait_asynccnt)
#define USE_ASYNC 1
#else
#define USE_ASYNC 0
#endif

// ---------------------------------------------------------------------------
// K0: fold final 1x1x1 conv (32x8) into the three 3x3x3 conv weights:
//     wcomb[s][o][k] = sum_g wfin[o][g] * w_s[g][gi][kd][kh][kw],
//     k = ((gi*3+kd)*3+kh)*3+kw.   Also emit packed per-k tap metadata.
// ---------------------------------------------------------------------------
__global__ void k_wcomb(const float* __restrict__ w_s,
                        const float* __restrict__ w_m,
                        const float* __restrict__ w_l,
                        const float* __restrict__ wfin,
                        float* __restrict__ wcomb,
                        int* __restrict__ kmeta) {
  int i = blockIdx.x * blockDim.x + threadIdx.x;
  if (i >= 3 * OUT * KTOT) return;
  int k = i % KTOT;
  int o = (i / KTOT) % OUT;
  int s = i / (KTOT * OUT);
  const float* wp = (s == 0) ? w_s : (s == 1) ? w_m : w_l;
  int gi = k / 27;
  int rem = k - gi * 27;          // (kd*3+kh)*3+kw
  float acc = 0.f;
#pragma unroll
  for (int g = 0; g < Gg; ++g)
    acc += wfin[o * Gg + g] * wp[(g * Gg + gi) * 27 + rem];
  wcomb[i] = acc;
  if (i < KTOT) {
    int kd = rem / 9;
    int r2 = rem - kd * 9;
    int kh = r2 / 3;
    int kw = r2 - kh * 3;
    kmeta[i] = gi | (kd << 4) | (kh << 6) | (kw << 8);
  }
}

// ---------------------------------------------------------------------------
// K1: per-group L2 normalization of feat_l and feat_r
// ---------------------------------------------------------------------------
__global__ void k_norm(const float* __restrict__ fl,
                       const float* __restrict__ fr,
                       float* __restrict__ fln,
                       float* __restrict__ frn) {
  int i = blockIdx.x * blockDim.x + threadIdx.x;
  const int total = Bb * Gg * Hh * Ww;
  if (i >= total) return;
  int w = i % Ww;
  int h = (i / Ww) % Hh;
  int g = (i / (Ww * Hh)) % Gg;
  int b = i / (Ww * Hh * Gg);
  const int plane = Hh * Ww;
  int off0 = ((b * Cc + g * GC) * Hh + h) * Ww + w;
#pragma unroll
  for (int t = 0; t < 2; ++t) {
    const float* src = t ? fr : fl;
    float* dst = t ? frn : fln;
    float ss = 0.f;
#pragma unroll
    for (int c = 0; c < GC; ++c) {
      float v = src[off0 + c * plane];
      ss += v * v;
    }
    float scale = 1.f / fmaxf(sqrtf(ss), 1e-12f);
#pragma unroll
    for (int c = 0; c < GC; ++c)
      dst[off0 + c * plane] = src[off0 + c * plane] * scale;
  }
}

// ---------------------------------------------------------------------------
// K2: correlation volume. One block per (b,g,h); stage the 8x192 fl/fr rows
// in LDS and reuse across all 48 disparities.
// ---------------------------------------------------------------------------
__global__ __launch_bounds__(256) void k_corr(const float* __restrict__ fln,
                                              const float* __restrict__ frn,
                                              float* __restrict__ base) {
  int idx = blockIdx.x;
  int h = idx % Hh;
  int g = (idx / Hh) % Gg;
  int b = idx / (Hh * Gg);
  __shared__ float sfl[GC * Ww];
  __shared__ float sfr[GC * Ww];
  for (int i = threadIdx.x; i < GC * Ww; i += blockDim.x) {
    int c = i / Ww, w = i % Ww;
    int off = ((b * Cc + g * GC + c) * Hh + h) * Ww + w;
    sfl[i] = fln[off];
    sfr[i] = frn[off];
  }
  __syncthreads();
  const float inv = 0.35355339059327373f;  // 1/sqrt(8)
  for (int i = threadIdx.x; i < Dd * Ww; i += blockDim.x) {
    int d = i / Ww, w = i % Ww;
    float acc = 0.f;
    int wr = w - d;
    if (wr >= 0) {
#pragma unroll
      for (int c = 0; c < GC; ++c) acc += sfl[c * Ww + w] * sfr[c * Ww + wr];
      acc *= inv;
    }
    base[(((b * Gg + g) * Dd + d) * Hh + h) * Ww + w] = acc;
  }
}

// ---------------------------------------------------------------------------
// K3a: 3x3 conv (65 -> 32) + BN + ReLU
// ---------------------------------------------------------------------------
__global__ void k_conv1(const float* __restrict__ feat_l,
                        const float* __restrict__ edge,
                        const float* __restrict__ w1,
                        const float* __restrict__ bn_g,
                        const float* __restrict__ bn_b,
                        const float* __restrict__ bn_m,
                        const float* __restrict__ bn_v,
                        float* __restrict__ x1) {
  int i = blockIdx.x * blockDim.x + threadIdx.x;
  const int total = Bb * 32 * Hh * Ww;
  if (i >= total) return;
  int w = i % Ww;
  int h = (i / Ww) % Hh;
  int oc = (i / (Ww * Hh)) % 32;
  int b = i / (Ww * Hh * 32);
  float acc = 0.f;
  for (int ic = 0; ic < 65; ++ic) {
    const float* plane = (ic < 64) ? (feat_l + (b * Cc + ic) * Hh * Ww)
                                   : (edge + b * Hh * Ww);
    const float* wp = w1 + (oc * 65 + ic) * 9;
#pragma unroll
    for (int kh = 0; kh < 3; ++kh) {
      int zh = h + kh - 1;
      if ((unsigned)zh >= (unsigned)Hh) continue;
#pragma unroll
      for (int kw = 0; kw < 3; ++kw) {
        int zw = w + kw - 1;
        if ((unsigned)zw >= (unsigned)Ww) continue;
        acc += plane[zh * Ww + zw] * wp[kh * 3 + kw];
      }
    }
  }
  float scale = bn_g[oc] * rsqrtf(bn_v[oc] + 1e-5f);
  acc = (acc - bn_m[oc]) * scale + bn_b[oc];
  x1[i] = fmaxf(acc, 0.f);
}

// ---------------------------------------------------------------------------
// K3b: 1x1 conv (32 -> 3) + temperature softmax -> per-pixel blend weights
// ---------------------------------------------------------------------------
__global__ void k_sw(const float* __restrict__ x1,
                     const float* __restrict__ w2,
                     const float* __restrict__ b2,
                     const float* __restrict__ temp,
                     float* __restrict__ swp) {
  int i = blockIdx.x * blockDim.x + threadIdx.x;
  const int total = Bb * Hh * Ww;
  if (i >= total) return;
  int w = i % Ww;
  int h = (i / Ww) % Hh;
  int b = i / (Ww * Hh);
  float lg[3];
#pragma unroll
  for (int s = 0; s < 3; ++s) {
    float acc = b2[s];
    for (int oc = 0; oc < 32; ++oc)
      acc += w2[s * 32 + oc] * x1[((b * 32 + oc) * Hh + h) * Ww + w];
    lg[s] = acc;
  }
  float t = fmaxf(temp[0], 0.1f);
  float m = fmaxf(lg[0], fmaxf(lg[1], lg[2]));
  float e0 = __expf((lg[0] - m) / t);
  float e1 = __expf((lg[1] - m) / t);
  float e2 = __expf((lg[2] - m) / t);
  float inv = 1.f / (e0 + e1 + e2);
  swp[((b * 3 + 0) * Hh + h) * Ww + w] = e0 * inv;
  swp[((b * 3 + 1) * Hh + h) * Ww + w] = e1 * inv;
  swp[((b * 3 + 2) * Hh + h) * Ww + w] = e2 * inv;
}

// ---------------------------------------------------------------------------
// Panel builder: fill one 216 x 64 im2col panel (row stride PSTR) for one
// dilation. Valid elements via async global->LDS DMA; padding via ds zeros.
// No divisions: tap decomposition comes from the precomputed kmeta table.
// ---------------------------------------------------------------------------
__device__ __forceinline__ void build_panel(float* pan,
                                            const float* __restrict__ basep,
                                            const int* __restrict__ kmeta,
                                            int bg, int d, int h, int w0,
                                            int dil, int tid) {
  for (int t = tid; t < KTOT * 64; t += 128) {
    int k = t >> 6;
    int wl = t & 63;
    int meta = kmeta[k];
    int gi = meta & 15;
    int dd = ((meta >> 4) & 3) - 1;
    int dh = ((meta >> 6) & 3) - 1;
    int dw = ((meta >> 8) & 3) - 1;
    int zd = d + dil * dd;
    int zh = h + dil * dh;
    int zw = w0 + wl + dil * dw;
    float* dst = pan + k * PSTR + wl;
    bool valid = (unsigned)zd < (unsigned)Dd && (unsigned)zh < (unsigned)Hh &&
                 (unsigned)zw < (unsigned)Ww;
#if USE_ASYNC
    if (valid) {
      const float* src = basep + (((bg + gi) * Dd + zd) * Hh + zh) * Ww + zw;
      // builtin takes generic (non-const) pointers: (src_global, dst_lds, off, cpol)
      __builtin_amdgcn_global_load_async_to_lds_b32((int*)src, (int*)dst, 0, 0);
    } else {
      *dst = 0.f;
    }
#else
    *dst = valid ? basep[(((bg + gi) * Dd + zd) * Hh + zh) * Ww + zw] : 0.f;
#endif
  }
}

// ---------------------------------------------------------------------------
// K4: fused dilated-conv + final-conv via fp32 WMMA.
// out[o,n] = sum_s sw_s[n] * (Wcomb_s @ im2col_s(base))[o,n] + bias[o]
// One block = 4 waves = 64 W-positions of one (b,d,h). Each wave: N-tile of
// 16 columns, two M-tiles (out channels 0-15 / 16-31), K=216 in steps of 4
// with V_WMMA_F32_16X16X4_F32. B-panels double-buffered in dynamic LDS and
// filled by the async DMA path while the previous panel is consumed.
// ---------------------------------------------------------------------------
__global__ __launch_bounds__(128) void k_main(const float* __restrict__ base,
                                              const float* __restrict__ wcomb,
                                              const int* __restrict__ kmeta,
                                              const float* __restrict__ swp,
                                              const float* __restrict__ bfin,
                                              float* __restrict__ out) {
  extern __shared__ float smem[];  // 2 * PANSZ floats

  int idx = blockIdx.x;
  int wt = idx % 3;
  int h = (idx / 3) % Hh;
  int d = (idx / (3 * Hh)) % Dd;
  int b = idx / (3 * Hh * Dd);
  int w0 = wt * 64;
  int bg = b * Gg;

  int tid = threadIdx.x;
  int wave = tid >> 5;
  int lane = tid & 31;
  int ln = lane & 15;   // N column / A row
  int hi = lane >> 4;   // selects K pair {0,1} vs {2,3}
  int wq = wave * 16;
  int w = w0 + wq + ln;

  const int dils[3] = {1, 2, 4};

  // prime buffer 0 with the dil=1 panel
  build_panel(smem, base, kmeta, bg, d, h, w0, 1, tid);

  v8f acc0 = {};
  v8f acc1 = {};

  for (int s = 0; s < 3; ++s) {
#if USE_ASYNC
    __builtin_amdgcn_s_wait_asynccnt(0);
#endif
    __syncthreads();
    if (s < 2)
      build_panel(smem + ((s + 1) & 1) * PANSZ, base, kmeta, bg, d, h, w0,
                  dils[s + 1], tid);

    const float* Wc = wcomb + s * OUT * KTOT;
    const float* pan = smem + (s & 1) * PANSZ;
    v8f c0 = {};
    v8f c1 = {};
    for (int kk = 0; kk < KTOT / 4; ++kk) {
      int k0 = kk * 4 + hi * 2;
      v2f a0, a1, bb;
      a0.x = Wc[ln * KTOT + k0];
      a0.y = Wc[ln * KTOT + k0 + 1];
      a1.x = Wc[(ln + 16) * KTOT + k0];
      a1.y = Wc[(ln + 16) * KTOT + k0 + 1];
      bb.x = pan[k0 * PSTR + wq + ln];
      bb.y = pan[(k0 + 1) * PSTR + wq + ln];
      c0 = __builtin_amdgcn_wmma_f32_16x16x4_f32(false, a0, false, bb,
                                                 (short)0, c0, false, false);
      c1 = __builtin_amdgcn_wmma_f32_16x16x4_f32(false, a1, false, bb,
                                                 (short)0, c1, false, false);
    }
    float swv = swp[((b * 3 + s) * Hh + h) * Ww + w];
#pragma unroll
    for (int v = 0; v < 8; ++v) {
      acc0[v] += c0[v] * swv;
      acc1[v] += c1[v] * swv;
    }
  }

#pragma unroll
  for (int v = 0; v < 8; ++v) {
    int oc0 = v + hi * 8;         // M-tile 0: channels 0..15
    int oc1 = 16 + v + hi * 8;    // M-tile 1: channels 16..31
    out[(((b * OUT + oc0) * Dd + d) * Hh + h) * Ww + w] = acc0[v] + bfin[oc0];
    out[(((b * OUT + oc1) * Dd + d) * Hh + h) * Ww + w] = acc1[v] + bfin[oc1];
  }
}

// ---------------------------------------------------------------------------
extern "C" void kernel_launch(void* const* d_in, const int* in_sizes, int n_in,
                              void* d_out, int out_size, void* d_ws, size_t ws_size,
                              hipStream_t stream) {
  const float* feat_l  = (const float*)d_in[0];
  const float* feat_r  = (const float*)d_in[1];
  const float* edge    = (const float*)d_in[2];
  const float* w_pred1 = (const float*)d_in[3];
  const float* bn_g    = (const float*)d_in[4];
  const float* bn_b    = (const float*)d_in[5];
  const float* bn_m    = (const float*)d_in[6];
  const float* bn_v    = (const float*)d_in[7];
  const float* w_pred2 = (const float*)d_in[8];
  const float* b_pred2 = (const float*)d_in[9];
  const float* temp    = (const float*)d_in[10];
  const float* w_s     = (const float*)d_in[11];
  const float* w_m     = (const float*)d_in[12];
  const float* w_l     = (const float*)d_in[13];
  const float* w_final = (const float*)d_in[14];
  const float* b_final = (const float*)d_in[15];
  float* out = (float*)d_out;

  // workspace layout (floats)
  float* ws    = (float*)d_ws;
  float* fln   = ws;
  float* frn   = fln  + (size_t)Bb * Cc * Hh * Ww;
  float* base  = frn  + (size_t)Bb * Cc * Hh * Ww;
  float* x1    = base + (size_t)Bb * Gg * Dd * Hh * Ww;
  float* swb   = x1   + (size_t)Bb * 32 * Hh * Ww;
  float* wcomb = swb  + (size_t)Bb * 3 * Hh * Ww;
  int*   kmeta = (int*)(wcomb + (size_t)3 * OUT * KTOT);

  {
    int n = 3 * OUT * KTOT;
    k_wcomb<<<(n + 255) / 256, 256, 0, stream>>>(w_s, w_m, w_l, w_final,
                                                 wcomb, kmeta);
  }
  {
    int n = Bb * Gg * Hh * Ww;
    k_norm<<<(n + 255) / 256, 256, 0, stream>>>(feat_l, feat_r, fln, frn);
  }
  k_corr<<<Bb * Gg * Hh, 256, 0, stream>>>(fln, frn, base);
  {
    int n = Bb * 32 * Hh * Ww;
    k_conv1<<<(n + 255) / 256, 256, 0, stream>>>(feat_l, edge, w_pred1,
                                                 bn_g, bn_b, bn_m, bn_v, x1);
  }
  {
    int n = Bb * Hh * Ww;
    k_sw<<<(n + 127) / 128, 128, 0, stream>>>(x1, w_pred2, b_pred2, temp, swb);
  }
  size_t shmem = (size_t)2 * PANSZ * sizeof(float);  // 124,416 B
  k_main<<<Bb * Dd * Hh * 3, 128, shmem, stream>>>(base, wcomb, kmeta, swb,
                                                   b_final, out);
}